// SFSAS_9423158248009
// MI455X (gfx1250) — compile-verified
//
#include <hip/hip_runtime.h>
#include <hip/hip_bf16.h>

typedef __attribute__((ext_vector_type(2))) float v2f;
typedef __attribute__((ext_vector_type(8))) float v8f;

// Set to 0 if the integrated assembler rejects the async mnemonic form.
#define USE_ASYNC_LDS 1

__device__ __forceinline__ v8f wmma_f32_4(v2f a, v2f b, v8f c) {
  // V_WMMA_F32_16X16X4_F32 : D = A(16x4) * B(4x16) + C(16x16), fp32
  return __builtin_amdgcn_wmma_f32_16x16x4_f32(
      /*neg_a=*/false, a, /*neg_b=*/false, b,
      /*c_mod=*/(short)0, c, /*reuse_a=*/false, /*reuse_b=*/false);
}

// ---------------------------------------------------------------------------
// 1) Global average pool over H*W for 16 channels: pooled[b*16+c]
//    b128 loads: 64 float4 per thread.
// ---------------------------------------------------------------------------
__global__ __launch_bounds__(256) void pool_mean(const float* __restrict__ in,
                                                 float* __restrict__ pooled,
                                                 int cb) {
  const int b = blockIdx.x >> 4;
  const int c = blockIdx.x & 15;
  const float4* p4 =
      (const float4*)(in + (((size_t)(b * cb + c)) << 16));
  float s = 0.f;
  for (int i = threadIdx.x; i < 16384; i += 256) {
    const float4 v = p4[i];
    s += (v.x + v.y) + (v.z + v.w);
  }
  __shared__ float red[256];
  red[threadIdx.x] = s;
  __syncthreads();
  for (int off = 128; off > 0; off >>= 1) {
    if (threadIdx.x < off) red[threadIdx.x] += red[threadIdx.x + off];
    __syncthreads();
  }
  if (threadIdx.x == 0) pooled[blockIdx.x] = red[0] * (1.f / 65536.f);
}

// ---------------------------------------------------------------------------
// 2) Tiny MLP: h = relu(pooled @ w1^T + b1); kern = h @ w2^T + b2
// ---------------------------------------------------------------------------
__global__ __launch_bounds__(256) void dyn_mlp(const float* __restrict__ pooled,
                                               const float* __restrict__ w1,
                                               const float* __restrict__ b1,
                                               const float* __restrict__ w2,
                                               const float* __restrict__ b2,
                                               float* __restrict__ kern,
                                               int OUT) {
  __shared__ float h[4][16];
  const int t = threadIdx.x;
  if (t < 64) {
    const int b = t >> 4, j = t & 15;
    float s = b1[j];
    for (int i = 0; i < 16; ++i) s += pooled[b * 16 + i] * w1[j * 16 + i];
    h[b][j] = s > 0.f ? s : 0.f;
  }
  __syncthreads();
  for (int idx = t; idx < 4 * OUT; idx += 256) {
    const int b = idx / OUT, o = idx % OUT;
    float s = b2[o];
    for (int j = 0; j < 16; ++j) s += h[b][j] * w2[o * 16 + j];
    kern[idx] = s;
  }
}

// ---------------------------------------------------------------------------
// 3) Dynamic depthwise KxK (cross-correlation, SAME/zero pad), per (b,c) kernel
// ---------------------------------------------------------------------------
__global__ __launch_bounds__(256) void dyn_dw(const float* __restrict__ in,
                                              const float* __restrict__ kern,
                                              float* __restrict__ out,
                                              int in_cb, int out_cb, int K) {
  const int b = blockIdx.y >> 4, c = blockIdx.y & 15;
  const int ty = (blockIdx.x >> 4) << 4, tx = (blockIdx.x & 15) << 4;
  __shared__ float kc[32];
  if (threadIdx.x < K * K)
    kc[threadIdx.x] = kern[(b * 16 + c) * K * K + threadIdx.x];
  __syncthreads();
  const int ly = threadIdx.x >> 4, lx = threadIdx.x & 15;
  const int y = ty + ly, x = tx + lx;
  const float* ip = in + (((size_t)(b * in_cb + c)) << 16);
  const int R = K >> 1;
  float s = 0.f;
  for (int dy = 0; dy < K; ++dy) {
    const int yy = y + dy - R;
    if ((unsigned)yy >= 256u) continue;
    for (int dx = 0; dx < K; ++dx) {
      const int xx = x + dx - R;
      if ((unsigned)xx >= 256u) continue;
      s += ip[(yy << 8) + xx] * kc[dy * K + dx];
    }
  }
  out[(((size_t)(b * out_cb + c)) << 16) + (y << 8) + x] = s;
}

// ---------------------------------------------------------------------------
// 4) Fused FSAS branch: per (batch, 8x8 patch) block.
//    conv1x1 48->144 (WMMA) -> depthwise 3x3 -> 8x8 circular conv (q (*) k)
//    -> channel LayerNorm * v -> conv1x1 48->48 (WMMA) -> concat ch 16..63
// ---------------------------------------------------------------------------
__global__ __launch_bounds__(256) void fsas_fused(
    const float* __restrict__ x, const float* __restrict__ hid_w,
    const float* __restrict__ hid_b, const float* __restrict__ dw_w,
    const float* __restrict__ dw_b, const float* __restrict__ ln_w,
    const float* __restrict__ ln_b, const float* __restrict__ out_w,
    const float* __restrict__ out_b, float* __restrict__ concat) {
  // Overlaid LDS: 30720 floats = 120 KB
  __shared__ __align__(16) float smem[30720];
  float(*s_x)[112] = (float(*)[112])smem;                  // 48x112 (ph 1-2)
  float(*s_hid)[112] = (float(*)[112])(smem + 5376);       // 144x112 (ph 2-3)
  float(*s_dw)[64] = (float(*)[64])(smem + 5376 + 16128);  // 144x64 (ph 3-5)
  float(*s_cc)[64] = (float(*)[64])smem;                   // 48x64 (ph 4-5)
  float(*s_m)[68] = (float(*)[68])(smem + 3072);           // 48x68 padded (ph 5-6)

  const int b = blockIdx.y;
  const int pY = blockIdx.x >> 5, pX = blockIdx.x & 31;
  const int py0 = pY << 3, px0 = pX << 3;
  const int tid = threadIdx.x;
  const int lane = tid & 31, wave = tid >> 5;
  const int l15 = lane & 15, hi = lane >> 4;

  if (tid < 144) __builtin_prefetch(&hid_w[tid * 48], 0, 1);

  // ---- load 48ch x 10x10 halo tile of x2 (zero OOB; cols 100..111 zero) ----
  for (int i = tid; i < 48 * 112; i += 256) {
    const int ch = i / 112, p = i % 112;
    float v = 0.f;
    if (p < 100) {
      const int hy = p / 10, hx = p % 10;
      const int gy = py0 + hy - 1, gx = px0 + hx - 1;
      if ((unsigned)gy < 256u && (unsigned)gx < 256u)
        v = x[(((size_t)(b * 64 + 16 + ch)) << 16) + (gy << 8) + gx];
    }
    s_x[ch][p] = v;
  }
  __syncthreads();

  // ---- hidden = hid_w(144x48) @ s_x(48x112) + hid_b  (WMMA f32, K=48) ----
  for (int t = wave; t < 63; t += 8) {
    const int mt = t / 7, nt = t % 7;
    const int mrow = mt << 4;
    v8f acc;
#pragma unroll
    for (int g = 0; g < 8; ++g) acc[g] = hid_b[mrow + g + 8 * hi];
    const int arow = (mrow + l15) * 48;
    const int ncol = (nt << 4) + l15;
#pragma unroll
    for (int kk = 0; kk < 12; ++kk) {
      const int k0 = kk * 4 + 2 * hi;
      v2f a, bb;
      a.x = hid_w[arow + k0];
      a.y = hid_w[arow + k0 + 1];
      bb.x = s_x[k0][ncol];
      bb.y = s_x[k0 + 1][ncol];
      acc = wmma_f32_4(a, bb, acc);
    }
    if (ncol < 100) {
#pragma unroll
      for (int g = 0; g < 8; ++g) s_hid[mrow + g + 8 * hi][ncol] = acc[g];
    }
  }
  __syncthreads();

  // ---- depthwise 3x3 on hidden (zero pad at image border) ----
  for (int i = tid; i < 144 * 64; i += 256) {
    const int c = i >> 6, p = i & 63;
    const int ly = p >> 3, lx = p & 7;
    float s = dw_b[c];
#pragma unroll
    for (int dy = 0; dy < 3; ++dy) {
      const int gy = py0 + ly + dy - 1;
      if ((unsigned)gy >= 256u) continue;
#pragma unroll
      for (int dx = 0; dx < 3; ++dx) {
        const int gx = px0 + lx + dx - 1;
        if ((unsigned)gx >= 256u) continue;
        s += s_hid[c][(ly + dy) * 10 + (lx + dx)] * dw_w[c * 9 + dy * 3 + dx];
      }
    }
    s_dw[c][p] = s;
  }
  __syncthreads();

  // ---- 8x8 circular convolution: out = irfft2(rfft2(q)*rfft2(k)) ----
  for (int i = tid; i < 48 * 64; i += 256) {
    const int c = i >> 6, p = i & 63;
    const int oi = p >> 3, oj = p & 7;
    float s = 0.f;
#pragma unroll
    for (int u = 0; u < 8; ++u) {
      const int ri = ((oi - u) & 7) << 3;
#pragma unroll
      for (int v = 0; v < 8; ++v)
        s += s_dw[c][(u << 3) + v] * s_dw[48 + c][ri + ((oj - v) & 7)];
    }
    s_cc[c][p] = s;
  }
  __syncthreads();

  // ---- channel LayerNorm (48 ch) then * v ----
  if (tid < 64) {
    float u = 0.f;
    for (int c = 0; c < 48; ++c) u += s_cc[c][tid];
    u *= (1.f / 48.f);
    float var = 0.f;
    for (int c = 0; c < 48; ++c) {
      const float d = s_cc[c][tid] - u;
      var += d * d;
    }
    var *= (1.f / 48.f);
    const float inv = rsqrtf(var + 1e-6f);
    for (int c = 0; c < 48; ++c) {
      const float xn = (s_cc[c][tid] - u) * inv;
      s_m[c][tid] = (ln_w[c] * xn + ln_b[c]) * s_dw[96 + c][tid];
    }
  }
  __syncthreads();

  // ---- out = out_w(48x48) @ s_m(48x64) + out_b  (WMMA f32, K=48) ----
  for (int t = wave; t < 12; t += 8) {
    const int mt = t >> 2, nt = t & 3;
    const int mrow = mt << 4;
    v8f acc;
#pragma unroll
    for (int g = 0; g < 8; ++g) acc[g] = out_b[mrow + g + 8 * hi];
    const int arow = (mrow + l15) * 48;
    const int ncol = (nt << 4) + l15;
#pragma unroll
    for (int kk = 0; kk < 12; ++kk) {
      const int k0 = kk * 4 + 2 * hi;
      v2f a, bb;
      a.x = out_w[arow + k0];
      a.y = out_w[arow + k0 + 1];
      bb.x = s_m[k0][ncol];
      bb.y = s_m[k0 + 1][ncol];
      acc = wmma_f32_4(a, bb, acc);
    }
    const int gy = py0 + (ncol >> 3), gx = px0 + (ncol & 7);
#pragma unroll
    for (int g = 0; g < 8; ++g) {
      const int ch = 16 + mrow + g + 8 * hi;
      concat[(((size_t)(b * 64 + ch)) << 16) + (gy << 8) + gx] = acc[g];
    }
  }
}

// ---------------------------------------------------------------------------
// 5) Final conv1x1 64->64 over concat buffer (WMMA f32, K=64)
//    128 pixels x 64 ch staged via async b128 copies to LDS (ASYNCcnt path);
//    rows padded to 132 floats to kill half-wave bank conflicts.
// ---------------------------------------------------------------------------
__global__ __launch_bounds__(256) void last_conv(const float* __restrict__ concat,
                                                 const float* __restrict__ w,
                                                 const float* __restrict__ bias,
                                                 float* __restrict__ out) {
  __shared__ __align__(16) float s_t[64][132];
  const int b = blockIdx.y;
  const int p0 = blockIdx.x << 7;
  const int tid = threadIdx.x, lane = tid & 31, wave = tid >> 5;
  const int l15 = lane & 15, hi = lane >> 4;

#if USE_ASYNC_LDS
  // 2048 16-byte chunks: 8 async copies per thread, tracked by ASYNCcnt.
  for (int i = tid; i < 64 * 32; i += 256) {
    const int ch = i >> 5, q = (i & 31) << 2;
    const float* g = concat + (((size_t)(b * 64 + ch)) << 16) + p0 + q;
    const unsigned lds_off = (unsigned)(uintptr_t)(&s_t[ch][q]);
    asm volatile("global_load_async_to_lds_b128 %0, %1, off"
                 :
                 : "v"(lds_off), "v"((unsigned long long)(uintptr_t)g)
                 : "memory");
  }
  asm volatile("s_wait_asynccnt 0x0" ::: "memory");
#else
  for (int i = tid; i < 64 * 32; i += 256) {
    const int ch = i >> 5, q = (i & 31) << 2;
    const float4 v =
        *(const float4*)(concat + (((size_t)(b * 64 + ch)) << 16) + p0 + q);
    *(float4*)&s_t[ch][q] = v;
  }
#endif
  __syncthreads();

  const int nb = (wave << 4) + l15;
  for (int mt = 0; mt < 4; ++mt) {
    const int mrow = mt << 4;
    v8f acc;
#pragma unroll
    for (int g = 0; g < 8; ++g) acc[g] = bias[mrow + g + 8 * hi];
    const int arow = (mrow + l15) << 6;
#pragma unroll
    for (int kk = 0; kk < 16; ++kk) {
      const int k0 = kk * 4 + 2 * hi;
      v2f a, bb;
      a.x = w[arow + k0];
      a.y = w[arow + k0 + 1];
      bb.x = s_t[k0][nb];
      bb.y = s_t[k0 + 1][nb];
      acc = wmma_f32_4(a, bb, acc);
    }
#pragma unroll
    for (int g = 0; g < 8; ++g)
      out[(((size_t)(b * 64 + mrow + g + 8 * hi)) << 16) + p0 + nb] = acc[g];
  }
}

// ---------------------------------------------------------------------------
extern "C" void kernel_launch(void* const* d_in, const int* in_sizes, int n_in,
                              void* d_out, int out_size, void* d_ws,
                              size_t ws_size, hipStream_t stream) {
  (void)in_sizes; (void)n_in; (void)out_size; (void)ws_size;
  const float* x       = (const float*)d_in[0];
  const float* dl1_w1  = (const float*)d_in[1];
  const float* dl1_b1  = (const float*)d_in[2];
  const float* dl1_w2  = (const float*)d_in[3];
  const float* dl1_b2  = (const float*)d_in[4];
  const float* dl2_w1  = (const float*)d_in[5];
  const float* dl2_b1  = (const float*)d_in[6];
  const float* dl2_w2  = (const float*)d_in[7];
  const float* dl2_b2  = (const float*)d_in[8];
  const float* hid_w   = (const float*)d_in[9];
  const float* hid_b   = (const float*)d_in[10];
  const float* dw_w    = (const float*)d_in[11];
  const float* dw_b    = (const float*)d_in[12];
  const float* ln_w    = (const float*)d_in[13];
  const float* ln_b    = (const float*)d_in[14];
  const float* out_w   = (const float*)d_in[15];
  const float* out_b   = (const float*)d_in[16];
  const float* last_w  = (const float*)d_in[17];
  const float* last_b  = (const float*)d_in[18];

  float* ws      = (float*)d_ws;
  float* concat  = ws;                        // 4*64*65536 = 16,777,216 floats
  float* y1      = ws + 16777216;             // 4*16*65536 =  4,194,304 floats
  float* pooled1 = y1 + 4194304;              // 64
  float* kern1   = pooled1 + 64;              // 576
  float* pooled2 = kern1 + 576;               // 64
  float* kern2   = pooled2 + 64;              // 1600

  // x1 branch: dynamic_local(k=3) then dynamic_local(k=5)
  pool_mean<<<64, 256, 0, stream>>>(x, pooled1, 64);
  dyn_mlp<<<1, 256, 0, stream>>>(pooled1, dl1_w1, dl1_b1, dl1_w2, dl1_b2, kern1, 144);
  dyn_dw<<<dim3(256, 64), 256, 0, stream>>>(x, kern1, y1, 64, 16, 3);
  pool_mean<<<64, 256, 0, stream>>>(y1, pooled2, 16);
  dyn_mlp<<<1, 256, 0, stream>>>(pooled2, dl2_w1, dl2_b1, dl2_w2, dl2_b2, kern2, 400);
  dyn_dw<<<dim3(256, 64), 256, 0, stream>>>(y1, kern2, concat, 16, 64, 5);

  // x2 branch: fused FSAS -> concat channels 16..63
  fsas_fused<<<dim3(32 * 32, 4), 256, 0, stream>>>(x, hid_w, hid_b, dw_w, dw_b,
                                                   ln_w, ln_b, out_w, out_b,
                                                   concat);

  // final conv1x1 64->64
  last_conv<<<dim3(512, 4), 256, 0, stream>>>(concat, last_w, last_b,
                                              (float*)d_out);
}